// LocallyConnected1D_72000831750854
// MI455X (gfx1250) — compile-verified
//
#include <hip/hip_runtime.h>

typedef __attribute__((ext_vector_type(2))) float v2f;
typedef __attribute__((ext_vector_type(8))) float v8f;

namespace {
constexpr int kInCh      = 128;
constexpr int kOutCh     = 128;
constexpr int kInLen     = 512;
constexpr int kKernel    = 8;
constexpr int kOutLen    = 505;                // (512 - 8) / 1 + 1
constexpr int kK         = kInCh * kKernel;    // 1024 (flattened reduction dim)
constexpr int kWStrideL  = kOutCh * kK;        // 131072 floats per output position
constexpr int kXStrideB  = kInCh * kInLen;     // 65536 floats per batch
constexpr int kOutStride = kOutCh * kOutLen;   // 64640 floats per batch
}

// One workgroup per output position l. 8 waves; wave w handles out_ch tile
// [16w, 16w+16). Each wave accumulates all 4 batch tiles (M=64) so the
// weight (B) fragment is register-reused 4x. fp32 WMMA keeps exact precision.
__global__ __launch_bounds__(256) void lc1d_wmma_f32(
    const float* __restrict__ x,     // [64][128][512]
    const float* __restrict__ w,     // [505][128][128][8]
    const float* __restrict__ bias,  // [128][505]
    float* __restrict__ out) {       // [64][128][505]
  const int l      = blockIdx.x;
  const int wave   = threadIdx.x >> 5;
  const int lane   = threadIdx.x & 31;
  const int half   = lane >> 4;   // selects K pair {0,1} vs {2,3} in A/B frags
  const int lane16 = lane & 15;   // M index for A, N index for B/C/D
  const int o      = wave * 16 + lane16;  // this lane's B-matrix column

  // Weight row for channel o at position l; k0 = 4*kc + 2*half is even, so
  // each B fragment is an 8B-aligned float2 along the contiguous K axis.
  const float* Wb = w + (size_t)l * kWStrideL + (size_t)o * kK + 2 * half;
  // x base for batch tile 0 (batch = mt*16 + lane16), tap offset 2*half.
  const float* xb = x + (size_t)lane16 * kXStrideB + l + 2 * half;

  v8f acc[4] = {v8f{}, v8f{}, v8f{}, v8f{}};

  #pragma unroll 2
  for (int i = 0; i < kInCh; ++i) {
    const float* wr = Wb + i * kKernel;
    v2f bf0 = *(const v2f*)(wr);       // K = 8i + 2h, 8i + 2h + 1
    v2f bf1 = *(const v2f*)(wr + 4);   // K = 8i + 4 + 2h, ...
    __builtin_prefetch(wr + 128, 0, 1);  // weight stream, ~512B ahead
    const float* xi = xb + i * kInLen;
    #pragma unroll
    for (int mt = 0; mt < 4; ++mt) {
      const float* xm = xi + (size_t)(mt * 16) * kXStrideB;
      v2f a0, a1;
      a0.x = xm[0]; a0.y = xm[1];      // taps 2h, 2h+1 of this octet
      a1.x = xm[4]; a1.y = xm[5];      // taps 4+2h, 5+2h
      acc[mt] = __builtin_amdgcn_wmma_f32_16x16x4_f32(
          false, a0, false, bf0, (short)0, acc[mt], false, false);
      acc[mt] = __builtin_amdgcn_wmma_f32_16x16x4_f32(
          false, a1, false, bf1, (short)0, acc[mt], false, false);
    }
  }

  // D layout: VGPR r -> M = half*8 + r, lane16 -> N. Add bias, store.
  const float bv = bias[(size_t)o * kOutLen + l];
  #pragma unroll
  for (int mt = 0; mt < 4; ++mt) {
    float* op = out + (size_t)(mt * 16 + half * 8) * kOutStride
                    + (size_t)o * kOutLen + l;
    #pragma unroll
    for (int r = 0; r < 8; ++r)
      op[(size_t)r * kOutStride] = acc[mt][r] + bv;
  }
}

extern "C" void kernel_launch(void* const* d_in, const int* /*in_sizes*/,
                              int /*n_in*/, void* d_out, int /*out_size*/,
                              void* /*d_ws*/, size_t /*ws_size*/,
                              hipStream_t stream) {
  const float* x  = (const float*)d_in[0];
  const float* w  = (const float*)d_in[1];
  const float* b  = (const float*)d_in[2];
  float* out      = (float*)d_out;
  lc1d_wmma_f32<<<dim3(kOutLen), dim3(256), 0, stream>>>(x, w, b, out);
}